// _AdjacencyMatrixUpdate_24051816857979
// MI455X (gfx1250) — compile-verified
//
#include <hip/hip_runtime.h>

typedef float v2f __attribute__((ext_vector_type(2)));
typedef float v8f __attribute__((ext_vector_type(8)));

#define BN_EPS 1e-3f
// Problem constants from the reference: B=4, N=512, F=64, C=129
#define NNODES 2048   // B*N
#define FDIM   64
#define CDIM   129

// ---------------------------------------------------------------------------
// Kernel 1: per-node channel reductions via V_WMMA_F32_16X16X4_F32 (exact f32)
//   Si[g] = sum_f w[1+f]  * relu(nf[g,f]*inv[1+f]  + cc[1+f])
//   Sj[g] = sum_f w[65+f] * relu(nf[g,f]*inv[65+f] + cc[65+f])
// One wave handles 16 nodes; A = 16x4 relu'd features, B = weights broadcast
// across all 16 columns, accumulated over 16 chunks of K=4.
// ---------------------------------------------------------------------------
__global__ void __launch_bounds__(128) node_sums_kernel(
    const float* __restrict__ nf,     // [NNODES, FDIM]
    const float* __restrict__ gamma,
    const float* __restrict__ beta,
    const float* __restrict__ mm,
    const float* __restrict__ mv,
    const float* __restrict__ w,
    const float* __restrict__ cb,
    float* __restrict__ Si,           // [NNODES]
    float* __restrict__ Sj,           // [NNODES]
    float* __restrict__ sc)           // [4] = {inv0, cc0, w0, convb}
{
    __shared__ float sInv[CDIM];
    __shared__ float sCc[CDIM];
    __shared__ float sW[CDIM];

    const int tid = threadIdx.x;
    for (int c = tid; c < CDIM; c += 128) {
        float iv = gamma[c] * rsqrtf(mv[c] + BN_EPS);
        sInv[c] = iv;
        sCc[c]  = beta[c] - mm[c] * iv;
        sW[c]   = w[c];
    }
    __syncthreads();

    if (blockIdx.x == 0 && tid == 0) {
        sc[0] = sInv[0];
        sc[1] = sCc[0];
        sc[2] = sW[0];
        sc[3] = cb[0];
    }

    const int lane = tid & 31;
    const int wave = (blockIdx.x * 128 + tid) >> 5;  // 0..127
    const int m    = lane & 15;                      // row within 16-node tile
    const int klo  = (lane >> 4) << 1;               // 0 (lanes 0-15) or 2 (lanes 16-31)
    const int g    = wave * 16 + m;                  // global node index 0..2047
    const float* nrow = nf + g * FDIM;

    v8f ci = {};
    v8f cj = {};

#pragma unroll
    for (int k = 0; k < 16; ++k) {
        const int k4 = k * 4 + klo;   // feature index of this lane's first A elem
        float x0 = nrow[k4];
        float x1 = nrow[k4 + 1];

        v2f ai, aj, bi, bj;
        // feat_i channels: c = 1 + f
        ai.x = fmaxf(fmaf(x0, sInv[1 + k4], sCc[1 + k4]), 0.0f);
        ai.y = fmaxf(fmaf(x1, sInv[2 + k4], sCc[2 + k4]), 0.0f);
        // feat_j channels: c = 65 + f
        aj.x = fmaxf(fmaf(x0, sInv[65 + k4], sCc[65 + k4]), 0.0f);
        aj.y = fmaxf(fmaf(x1, sInv[66 + k4], sCc[66 + k4]), 0.0f);
        // B: same weight for every column N -> every output column = row sum
        bi.x = sW[1 + k4];
        bi.y = sW[2 + k4];
        bj.x = sW[65 + k4];
        bj.y = sW[66 + k4];

        ci = __builtin_amdgcn_wmma_f32_16x16x4_f32(false, ai, false, bi,
                                                   (short)0, ci, false, false);
        cj = __builtin_amdgcn_wmma_f32_16x16x4_f32(false, aj, false, bj,
                                                   (short)0, cj, false, false);
    }

    // D layout (16x16 f32): lane L holds column N=L%16; VGPR r = row r (lanes
    // 0-15) or row r+8 (lanes 16-31). Column 0 lives in lanes 0 and 16.
    if ((lane & 15) == 0) {
        const int rbase = wave * 16 + (lane >> 4) * 8;
#pragma unroll
        for (int r = 0; r < 8; ++r) {
            Si[rbase + r] = ci[r];
            Sj[rbase + r] = cj[r];
        }
    }
}

// ---------------------------------------------------------------------------
// Kernel 2: streaming fusion, the bandwidth-critical part (8.5 MiB traffic).
//   out[b,i,j] = w0*relu(adj*inv0 + c0) + Si[b*N+i] + Sj[b*N+j] + cb
// One float4 (4 consecutive j) per thread.
// ---------------------------------------------------------------------------
__global__ void __launch_bounds__(256) fuse_kernel(
    const float4* __restrict__ adj4,  // [B*N*N/4]
    const float*  __restrict__ Si,    // [NNODES]
    const float4* __restrict__ Sj4,   // [NNODES/4] viewed as float4
    const float*  __restrict__ sc,    // {inv0, c0, w0, cb}
    float4* __restrict__ out4)
{
    const int idx = blockIdx.x * 256 + threadIdx.x;   // 0..262143
    const float inv0 = sc[0];
    const float c0   = sc[1];
    const float w0   = sc[2];
    const float cb   = sc[3];

    const int r  = idx >> 7;     // global row b*N+i   (N/4 = 128)
    const int j4 = idx & 127;
    const int b  = r >> 9;       // /512

    float4 a  = adj4[idx];
    float  si = Si[r] + cb;
    float4 sj = Sj4[b * 128 + j4];

    float4 o;
    o.x = fmaf(fmaxf(fmaf(a.x, inv0, c0), 0.0f), w0, si + sj.x);
    o.y = fmaf(fmaxf(fmaf(a.y, inv0, c0), 0.0f), w0, si + sj.y);
    o.z = fmaf(fmaxf(fmaf(a.z, inv0, c0), 0.0f), w0, si + sj.z);
    o.w = fmaf(fmaxf(fmaf(a.w, inv0, c0), 0.0f), w0, si + sj.w);
    out4[idx] = o;
}

extern "C" void kernel_launch(void* const* d_in, const int* in_sizes, int n_in,
                              void* d_out, int out_size, void* d_ws, size_t ws_size,
                              hipStream_t stream) {
    const float* nf    = (const float*)d_in[0];  // node_features [4,512,64]
    const float* adj   = (const float*)d_in[1];  // adjacency    [4,512,512]
    const float* gamma = (const float*)d_in[2];  // [129]
    const float* beta  = (const float*)d_in[3];  // [129]
    const float* mm    = (const float*)d_in[4];  // [129]
    const float* mv    = (const float*)d_in[5];  // [129]
    const float* w     = (const float*)d_in[6];  // [129]
    const float* cb    = (const float*)d_in[7];  // [1]

    float* ws = (float*)d_ws;
    float* Si = ws;              // 2048 floats
    float* Sj = ws + NNODES;     // 2048 floats
    float* sc = ws + 2 * NNODES; // 4 floats

    // 128 waves * 16 nodes = 2048 nodes
    node_sums_kernel<<<32, 128, 0, stream>>>(nf, gamma, beta, mm, mv, w, cb,
                                             Si, Sj, sc);
    // 4*512*512/4 = 262144 float4s
    fuse_kernel<<<1024, 256, 0, stream>>>((const float4*)adj, Si,
                                          (const float4*)Sj, sc,
                                          (float4*)d_out);
}